// TreeCRFLayer_89189290869443
// MI455X (gfx1250) — compile-verified
//
#include <hip/hip_runtime.h>
#include <cstdint>

#define NLV 11
#define L   2047          // number of tree nodes
#define CPAD 2048         // padded edge-table stride (indexed by child id)

// stable 2-way logsumexp
__device__ __forceinline__ float lse2(float a, float b) {
    float m = fmaxf(a, b);
    float d = fabsf(a - b);
    return m + log1pf(expf(-d));
}

// generic pointer -> LDS byte offset (for async_to_lds VDST operand)
__device__ __forceinline__ unsigned lds_off(const void* p) {
    return (unsigned)(unsigned long long)
           (__attribute__((address_space(3))) const void*)p;
}

__global__ __launch_bounds__(256)
void treecrf_kernel(const float* __restrict__ em,
                    const float4* __restrict__ tup,
                    const float4* __restrict__ tdn,
                    float* __restrict__ out) {
    __shared__ __align__(16) float sE[2 * L];   // staged emissions  [c*L + node]
    __shared__ __align__(16) float sA[2 * L];   // upward messages  (alphas)
    __shared__ __align__(16) float sB[2 * L];   // downward messages (betas)

    const int b    = blockIdx.x;
    const int tid  = threadIdx.x;
    const int nthr = blockDim.x;

    // ---- async DMA: emissions[b] (2*L floats, contiguous, 8B aligned) -> LDS
    const unsigned long long gbase =
        (unsigned long long)(const void*)(em + (size_t)b * (2 * L));
    const unsigned lbase = lds_off(sE);
    for (int i = tid; i < L; i += nthr) {                 // L chunks of 8 bytes
        unsigned lo = lbase + (unsigned)(i << 3);
        unsigned go = (unsigned)(i << 3);
        asm volatile("global_load_async_to_lds_b64 %0, %1, %2"
                     :: "v"(lo), "v"(go), "s"(gbase) : "memory");
    }
    // overlap with the DMA: warm the edge tables, zero alphas, init root beta
    __builtin_prefetch((const char*)tup + (size_t)tid * 128, 0, 1);
    __builtin_prefetch((const char*)tdn + (size_t)tid * 128, 0, 1);
    for (int i = tid; i < 2 * L; i += nthr) sA[i] = 0.0f;
    if (tid == 0) { sB[0] = 0.0f; sB[L] = 0.0f; }
    asm volatile("s_wait_asynccnt 0x0" ::: "memory");
    __syncthreads();

    // ---- upward sweep: leaves -> root (each thread owns one parent)
    for (int lev = NLV - 1; lev >= 1; --lev) {
        int np = 1 << (lev - 1);
        int pfirst = np - 1;
        for (int j = tid; j < np; j += nthr) {
            int pp = pfirst + j;
            int cl = 2 * pp + 1, cr = 2 * pp + 2;
            float a0, a1;
            {
                float s0 = sE[cl] + sA[cl];
                float s1 = sE[L + cl] + sA[L + cl];
                float4 T = tup[cl];               // x=T00 y=T01 z=T10 w=T11
                a0 = lse2(s0 + T.x, s1 + T.y);
                a1 = lse2(s0 + T.z, s1 + T.w);
            }
            {
                float s0 = sE[cr] + sA[cr];
                float s1 = sE[L + cr] + sA[L + cr];
                float4 T = tup[cr];
                a0 += lse2(s0 + T.x, s1 + T.y);
                a1 += lse2(s0 + T.z, s1 + T.w);
            }
            sA[pp]     = a0;
            sA[L + pp] = a1;
        }
        __syncthreads();
    }

    // ---- downward sweep: root -> leaves (each thread owns one child)
    for (int lev = 1; lev < NLV; ++lev) {
        int nc = 1 << lev;
        int cfirst = nc - 1;
        for (int j = tid; j < nc; j += nthr) {
            int ch = cfirst + j;
            int pp = (ch - 1) >> 1;
            float s0 = sE[pp]     + sB[pp];
            float s1 = sE[L + pp] + sB[L + pp];
            float4 T = tdn[ch];
            sB[ch]     = lse2(s0 + T.x, s1 + T.y);
            sB[L + ch] = lse2(s0 + T.z, s1 + T.w);
        }
        __syncthreads();
    }

    // ---- scores = E + A + B, log-softmax over the 2 classes, stream out
    float* outb = out + (size_t)b * (2 * L);
    for (int l = tid; l < L; l += nthr) {
        float s0 = sE[l]     + sA[l]     + sB[l];
        float s1 = sE[L + l] + sA[L + l] + sB[L + l];
        float z  = lse2(s0, s1);
        outb[l]     = s0 - z;
        outb[L + l] = s1 - z;
    }
}

// Gather the only transition blocks ever used: for each child i (1..2046)
//   tup[i] = transitions[parent(i), i]   (2x2, upward edge)
//   tdn[i] = transitions[i, parent(i)]   (2x2, downward edge)
__global__ void compact_trans_kernel(const float4* __restrict__ t4,
                                     float4* __restrict__ tup,
                                     float4* __restrict__ tdn) {
    int i = blockIdx.x * blockDim.x + threadIdx.x + 1;   // child ids 1..L-1
    if (i >= L) return;
    int p = (i - 1) >> 1;
    tup[i] = t4[(size_t)p * L + i];
    tdn[i] = t4[(size_t)i * L + p];
}

extern "C" void kernel_launch(void* const* d_in, const int* in_sizes, int n_in,
                              void* d_out, int out_size, void* d_ws, size_t ws_size,
                              hipStream_t stream) {
    const float*  em = (const float*)d_in[0];                  // [B, 2, 2047]
    const float4* tr = (const float4*)d_in[1];                 // [2047,2047] x (2x2)
    float*  out = (float*)d_out;
    float4* tup = (float4*)d_ws;                               // 2048 entries
    float4* tdn = tup + CPAD;                                  // 2048 entries (64 KB total)

    int batch = in_sizes[0] / (2 * L);

    compact_trans_kernel<<<(L + 255) / 256, 256, 0, stream>>>(tr, tup, tdn);
    treecrf_kernel<<<batch, 256, 0, stream>>>(em, tup, tdn, out);
}